// CNNNet_DGLNetwork_18150531793006
// MI455X (gfx1250) — compile-verified
//
#include <hip/hip_runtime.h>
#include <hip/hip_bf16.h>

typedef __attribute__((ext_vector_type(2))) float v2f;
typedef __attribute__((ext_vector_type(8))) float v8f;

#define N_NODES 100000
#define N_EDGES 3200000
#define IN_FEATS 1433

// ---------------- utility kernels ----------------

__global__ void gcn_zero_f32(float* __restrict__ p, long long n) {
    long long i = (long long)blockIdx.x * blockDim.x + threadIdx.x;
    long long stride = (long long)gridDim.x * blockDim.x;
    for (; i < n; i += stride) p[i] = 0.0f;
}

__global__ void gcn_degrees(const long long* __restrict__ edges,
                            float* __restrict__ deg_out, float* __restrict__ deg_in,
                            int nedges) {
    int e = blockIdx.x * blockDim.x + threadIdx.x;
    if (e < nedges) {
        int s = (int)edges[e];
        int d = (int)edges[nedges + e];
        atomicAdd(&deg_out[s], 1.0f);
        atomicAdd(&deg_in[d], 1.0f);
    }
}

__global__ void gcn_rsqrt_clamp(float* __restrict__ p, int n) {
    int i = blockIdx.x * blockDim.x + threadIdx.x;
    if (i < n) p[i] = rsqrtf(fmaxf(p[i], 1.0f));
}

// ---------------- WMMA GEMM: C[M x 16] = (rowscale .* A[M x K]) @ W[K x 16] ----------------
// One wave (32 lanes) per 16-row tile. fp32 WMMA 16x16x4, f32 accumulate.
// A-frag: lane l -> row m = tile*16 + (l&15), K pair {k0+2*hi, k0+2*hi+1}, hi = l>>4.
// B-frag: lane l -> col n = l&15, same K pair.
// C/D:    vgpr v -> row tile*16 + v + 8*hi, col n.
__global__ void gcn_gemm_wmma(const float* __restrict__ A,
                              const float* __restrict__ W,
                              const float* __restrict__ rowscale,
                              float* __restrict__ C,
                              int Mtiles, int K, int lda) {
    int wave = blockIdx.x * (blockDim.x >> 5) + (threadIdx.x >> 5);
    if (wave >= Mtiles) return;               // wave-uniform: EXEC stays all-1s for WMMA
    int lane = threadIdx.x & 31;
    int hi   = lane >> 4;                     // 0 or 1
    int n    = lane & 15;                     // A row within tile == B/C column
    long long m = (long long)wave * 16 + n;

    float ns = (rowscale != nullptr) ? rowscale[m] : 1.0f;
    const float* arow = A + m * (long long)lda;

    v8f c = {};
    int Kfull = K & ~3;
    int k0 = 0;
    #pragma unroll 4
    for (; k0 < Kfull; k0 += 4) {
        int k = k0 + 2 * hi;
        v2f a;
        a.x = arow[k]     * ns;
        a.y = arow[k + 1] * ns;
        v2f b;
        b.x = W[(long long)k * 16 + n];
        b.y = W[(long long)(k + 1) * 16 + n];
        c = __builtin_amdgcn_wmma_f32_16x16x4_f32(false, a, false, b,
                                                  (short)0, c, false, false);
    }
    if (k0 < K) {                             // masked K tail (K=1433 case)
        int k = k0 + 2 * hi;
        v2f a = {0.0f, 0.0f};
        v2f b = {0.0f, 0.0f};
        if (k < K)     { a.x = arow[k]     * ns; b.x = W[(long long)k * 16 + n]; }
        if (k + 1 < K) { a.y = arow[k + 1] * ns; b.y = W[(long long)(k + 1) * 16 + n]; }
        c = __builtin_amdgcn_wmma_f32_16x16x4_f32(false, a, false, b,
                                                  (short)0, c, false, false);
    }

    float* crow = C + ((long long)wave * 16 + 8 * hi) * 16 + n;
    #pragma unroll
    for (int v = 0; v < 8; ++v) crow[(long long)v * 16] = c[v];
}

// ---------------- edge scatter (segment_sum) ----------------

__global__ void gcn_scatter16(const long long* __restrict__ edges,
                              const float* __restrict__ x, float* __restrict__ m,
                              int nedges) {
    long long tid = (long long)blockIdx.x * blockDim.x + threadIdx.x;
    if (tid >= (long long)nedges * 16) return;
    int e = (int)(tid >> 4);
    int f = (int)(tid & 15);
    int s = (int)edges[e];
    int d = (int)edges[nedges + e];
    atomicAdd(&m[(long long)d * 16 + f], x[(long long)s * 16 + f]);
}

__global__ void gcn_scatter7(const long long* __restrict__ edges,
                             const float* __restrict__ x, float* __restrict__ out,
                             int nedges) {
    long long tid = (long long)blockIdx.x * blockDim.x + threadIdx.x;
    if (tid >= (long long)nedges * 8) return;
    int e = (int)(tid >> 3);
    int f = (int)(tid & 7);
    if (f >= 7) return;
    int s = (int)edges[e];
    int d = (int)edges[nedges + e];
    atomicAdd(&out[(long long)d * 7 + f], x[(long long)s * 16 + f]);
}

// h = relu(m*norm_dst + b1); store h*norm_src in place (pre-scaled for layer-2 GEMM)
__global__ void gcn_post1(float* __restrict__ m,
                          const float* __restrict__ nd, const float* __restrict__ ns,
                          const float* __restrict__ b1, int nnodes) {
    long long tid = (long long)blockIdx.x * blockDim.x + threadIdx.x;
    if (tid >= (long long)nnodes * 16) return;
    int node = (int)(tid >> 4);
    int f    = (int)(tid & 15);
    float v = fmaf(m[tid], nd[node], b1[f]);
    v = v > 0.0f ? v : 0.0f;
    m[tid] = v * ns[node];
}

__global__ void gcn_pack_w2(const float* __restrict__ W2, float* __restrict__ w2p) {
    int tid = threadIdx.x;                    // 256 threads, one block
    if (tid < 256) {
        int k = tid >> 4;
        int n = tid & 15;
        w2p[tid] = (n < 7) ? W2[k * 7 + n] : 0.0f;
    }
}

__global__ void gcn_final(float* __restrict__ out, const float* __restrict__ nd,
                          const float* __restrict__ b2, int nnodes) {
    long long tid = (long long)blockIdx.x * blockDim.x + threadIdx.x;
    if (tid >= (long long)nnodes * 7) return;
    int node = (int)(tid / 7);
    int f    = (int)(tid - (long long)node * 7);
    out[tid] = fmaf(out[tid], nd[node], b2[f]);
}

// ---------------- launch ----------------

extern "C" void kernel_launch(void* const* d_in, const int* in_sizes, int n_in,
                              void* d_out, int out_size, void* d_ws, size_t ws_size,
                              hipStream_t stream) {
    const float*     features = (const float*)d_in[0];
    const long long* edges    = (const long long*)d_in[1];
    const float*     W1       = (const float*)d_in[2];
    const float*     b1       = (const float*)d_in[3];
    const float*     W2       = (const float*)d_in[4];
    const float*     b2       = (const float*)d_in[5];
    float* out = (float*)d_out;
    float* ws  = (float*)d_ws;

    const int N = N_NODES, E = N_EDGES, K1 = IN_FEATS;
    const int Mtiles = N / 16;                 // 6250, exact

    float* norm_src = ws;                      // N floats (deg_out -> rsqrt in place)
    float* norm_dst = ws + N;                  // N floats
    float* xw1      = ws + 2 * N;              // N*16 floats (reused as x2 for layer 2)
    float* m1       = xw1 + (long long)N * 16; // N*16 floats (h, then pre-scaled h)
    float* w2p      = m1  + (long long)N * 16; // 256 floats

    // degrees -> inverse-sqrt norms
    gcn_zero_f32<<<512, 256, 0, stream>>>(norm_src, 2LL * N);
    gcn_degrees<<<(E + 255) / 256, 256, 0, stream>>>(edges, norm_src, norm_dst, E);
    gcn_rsqrt_clamp<<<(2 * N + 255) / 256, 256, 0, stream>>>(norm_src, 2 * N);

    // layer 1: xw1 = (X .* norm_src) @ W1   (WMMA fp32)
    gcn_gemm_wmma<<<(Mtiles + 7) / 8, 256, 0, stream>>>(features, W1, norm_src, xw1,
                                                        Mtiles, K1, K1);
    // m1 = segment_sum(xw1[src], dst)
    gcn_zero_f32<<<2048, 256, 0, stream>>>(m1, (long long)N * 16);
    gcn_scatter16<<<(int)(((long long)E * 16 + 255) / 256), 256, 0, stream>>>(edges, xw1, m1, E);
    // m1 = relu(m1*norm_dst + b1) * norm_src
    gcn_post1<<<(int)(((long long)N * 16 + 255) / 256), 256, 0, stream>>>(m1, norm_dst, norm_src, b1, N);

    // layer 2: x2(=xw1) = m1 @ W2pad  (WMMA fp32, K=16)
    gcn_pack_w2<<<1, 256, 0, stream>>>(W2, w2p);
    gcn_gemm_wmma<<<(Mtiles + 7) / 8, 256, 0, stream>>>(m1, w2p, nullptr, xw1,
                                                        Mtiles, 16, 16);
    // out = segment_sum(x2[src], dst) * norm_dst + b2
    gcn_zero_f32<<<1024, 256, 0, stream>>>(out, (long long)N * 7);
    gcn_scatter7<<<(int)(((long long)E * 8 + 255) / 256), 256, 0, stream>>>(edges, xw1, out, E);
    gcn_final<<<(int)(((long long)N * 7 + 255) / 256), 256, 0, stream>>>(out, norm_dst, b2, N);
}